// GraphModule_68066641707589
// MI455X (gfx1250) — compile-verified
//
#include <hip/hip_runtime.h>
#include <hip/hip_bf16.h>

typedef __attribute__((ext_vector_type(2))) float v2f;
typedef __attribute__((ext_vector_type(8))) float v8f;

#define D_FEAT 64
#define N_NODES_K 100000

// ---------------------------------------------------------------------------
// Phase 0: zero the aggregation buffer (which is d_out, used in-place).
// ---------------------------------------------------------------------------
__global__ void zero_kernel(float4* __restrict__ p, int n4) {
    int i = blockIdx.x * blockDim.x + threadIdx.x;
    if (i < n4) p[i] = make_float4(0.f, 0.f, 0.f, 0.f);
}

// ---------------------------------------------------------------------------
// Phase 1: msg = h[src] * w ; agg[dst] += msg   (edge-parallel scatter-add)
// 16 threads per edge, each owns a float4 chunk -> coalesced 256B row reads,
// 4 global f32 atomics per thread. agg (25.6MB) + h (25.6MB) are L2-resident.
// ---------------------------------------------------------------------------
__global__ void scatter_kernel(const float* __restrict__ h,
                               const float* __restrict__ ew,
                               const int* __restrict__ src,
                               const int* __restrict__ dst,
                               float* __restrict__ agg,
                               int E) {
    int t = blockIdx.x * blockDim.x + threadIdx.x;
    int e = t >> 4;          // edge index
    int c = t & 15;          // float4 chunk within the 64-float row
    if (e >= E) return;
    int s = src[e];
    int d = dst[e];
    float w = ew[e];
    const float4 v = *(const float4*)(h + (size_t)s * D_FEAT + c * 4);
    float* o = agg + (size_t)d * D_FEAT + c * 4;
    atomicAdd(o + 0, v.x * w);
    atomicAdd(o + 1, v.y * w);
    atomicAdd(o + 2, v.z * w);
    atomicAdd(o + 3, v.w * w);
}

// ---------------------------------------------------------------------------
// Phase 2: out = agg @ W^T + b, in-place in `io` ([N,64] f32).
// One wave32 per 16-row strip. V_WMMA_F32_16X16X4_F32, K looped 16x,
// 4 N-tiles of 16 accumulated simultaneously (4 x v8f accumulators).
// A(16x4) lane layout: lane<16 -> M=lane, K=k0..k0+1 ; lane>=16 -> K=k0+2..k0+3
// B(4x16): same K split, N = lane&15 ; B(k,j) = W[j,k] (row-major W, contiguous)
// D(16x16): VGPR r -> row mBase + r + 8*(lane>>4), col = lane&15
// ---------------------------------------------------------------------------
__global__ void gemm_wmma_kernel(float* __restrict__ io,
                                 const float* __restrict__ W,
                                 const float* __restrict__ bias,
                                 int N) {
    const int lane  = threadIdx.x & 31;
    const int wave  = threadIdx.x >> 5;
    const int gwave = blockIdx.x * (blockDim.x >> 5) + wave;
    const int mBase = gwave * 16;
    if (mBase >= N) return;              // wave-uniform exit: EXEC stays all-1s

    const int half = lane >> 4;          // 0: K pair {0,1}; 1: K pair {2,3}
    const int l    = lane & 15;

    v8f acc0 = {}, acc1 = {}, acc2 = {}, acc3 = {};

    const float* aRow = io + (size_t)(mBase + l) * D_FEAT;
    const float* w0   = W  + (size_t)(0 * 16 + l) * D_FEAT;
    const float* w1   = W  + (size_t)(1 * 16 + l) * D_FEAT;
    const float* w2   = W  + (size_t)(2 * 16 + l) * D_FEAT;
    const float* w3   = W  + (size_t)(3 * 16 + l) * D_FEAT;

#pragma unroll
    for (int kc = 0; kc < 16; ++kc) {
        const int k0 = kc * 4 + half * 2;          // even -> 8B aligned
        v2f a  = *(const v2f*)(aRow + k0);
        v2f b0 = *(const v2f*)(w0 + k0);
        v2f b1 = *(const v2f*)(w1 + k0);
        v2f b2 = *(const v2f*)(w2 + k0);
        v2f b3 = *(const v2f*)(w3 + k0);
        acc0 = __builtin_amdgcn_wmma_f32_16x16x4_f32(false, a, false, b0, (short)0, acc0, false, false);
        acc1 = __builtin_amdgcn_wmma_f32_16x16x4_f32(false, a, false, b1, (short)0, acc1, false, false);
        acc2 = __builtin_amdgcn_wmma_f32_16x16x4_f32(false, a, false, b2, (short)0, acc2, false, false);
        acc3 = __builtin_amdgcn_wmma_f32_16x16x4_f32(false, a, false, b3, (short)0, acc3, false, false);
    }

    const float bv0 = bias[ 0 + l];
    const float bv1 = bias[16 + l];
    const float bv2 = bias[32 + l];
    const float bv3 = bias[48 + l];

#pragma unroll
    for (int r = 0; r < 8; ++r) {
        const int row = mBase + r + half * 8;
        float* o = io + (size_t)row * D_FEAT;
        o[ 0 + l] = acc0[r] + bv0;
        o[16 + l] = acc1[r] + bv1;
        o[32 + l] = acc2[r] + bv2;
        o[48 + l] = acc3[r] + bv3;
    }
}

// ---------------------------------------------------------------------------
extern "C" void kernel_launch(void* const* d_in, const int* in_sizes, int n_in,
                              void* d_out, int out_size, void* d_ws, size_t ws_size,
                              hipStream_t stream) {
    const float* h  = (const float*)d_in[0];   // [N, 64]
    const float* ew = (const float*)d_in[1];   // [E, 1]
    const int*   src = (const int*)d_in[2];    // [E]
    const int*   dst = (const int*)d_in[3];    // [E]
    const float* W  = (const float*)d_in[4];   // [64, 64]
    const float* b  = (const float*)d_in[5];   // [64]

    const int E = in_sizes[2];
    const int N = N_NODES_K;

    float* out = (float*)d_out;                // used as agg, then overwritten in-place

    // Phase 0: zero agg (== d_out)
    {
        int n4 = (N * D_FEAT) / 4;
        int blocks = (n4 + 255) / 256;
        zero_kernel<<<blocks, 256, 0, stream>>>((float4*)out, n4);
    }
    // Phase 1: edge scatter-add
    {
        long long threads = (long long)E * 16;
        int blocks = (int)((threads + 255) / 256);
        scatter_kernel<<<blocks, 256, 0, stream>>>(h, ew, src, dst, out, E);
    }
    // Phase 2: WMMA GEMM + bias, in-place
    {
        int waves  = (N + 15) / 16;            // 6250
        int blocks = (waves + 7) / 8;          // 8 waves (256 threads) per block
        gemm_wmma_kernel<<<blocks, 256, 0, stream>>>(out, W, b, N);
    }
}